// GetWeightMatrix_25812753449084
// MI455X (gfx1250) — compile-verified
//
#include <hip/hip_runtime.h>
#include <hip/hip_bf16.h>
#include <cstdint>

// Problem constants (match reference): B=16, C=3, W=H=512, K=5, PAD=2, CENTER=12
#define IMG   512
#define TILE  32
#define HALO  2
#define LDIM  (TILE + 2 * HALO)   // 36
#define LSZ   (LDIM * LDIM)       // 1296
#define NTHREADS 256

__global__ __launch_bounds__(NTHREADS)
void stencil_wx_wy_kernel(const float* __restrict__ xin,
                          const float* __restrict__ yin,
                          float* __restrict__ out,
                          unsigned long long nelem)
{
    __shared__ float sx[LSZ];
    __shared__ float sy[LSZ];

    const int tx    = blockIdx.x;          // tile col  (0..15)
    const int ty    = blockIdx.y;          // tile row  (0..15)
    const int plane = blockIdx.z;          // B*C plane (0..47)
    const int tid   = threadIdx.y * 32 + threadIdx.x;

    const long pbase = (long)plane * (long)(IMG * IMG);
    const int  x0 = tx * TILE - HALO;      // top-left of halo'd input tile
    const int  y0 = ty * TILE - HALO;

    const bool interior = (tx >= 1) && (tx <= 14) && (ty >= 1) && (ty <= 14);

    if (interior) {
        // CDNA5 async DMA: global -> LDS, no VGPR staging. Tracked by ASYNCcnt.
        const uint32_t sxb = (uint32_t)(uintptr_t)(&sx[0]);  // low 32b of generic ptr == LDS byte addr
        const uint32_t syb = (uint32_t)(uintptr_t)(&sy[0]);
        for (int i = tid; i < LSZ; i += NTHREADS) {
            const int  ly = i / LDIM;
            const int  lc = i - ly * LDIM;
            const long g  = pbase + (long)(y0 + ly) * IMG + (x0 + lc);
            const uint64_t ga = (uint64_t)(uintptr_t)(xin + g);
            const uint64_t gb = (uint64_t)(uintptr_t)(yin + g);
            const uint32_t la = sxb + (uint32_t)i * 4u;
            const uint32_t lb = syb + (uint32_t)i * 4u;
            asm volatile("global_load_async_to_lds_b32 %0, %1, off"
                         :: "v"(la), "v"(ga) : "memory");
            asm volatile("global_load_async_to_lds_b32 %0, %1, off"
                         :: "v"(lb), "v"(gb) : "memory");
        }
        asm volatile("s_wait_asynccnt 0" ::: "memory");
    } else {
        // Border tiles: bounds-checked synchronous fill, zero halo (jnp.pad semantics).
        for (int i = tid; i < LSZ; i += NTHREADS) {
            const int ly = i / LDIM;
            const int lc = i - ly * LDIM;
            const int gy = y0 + ly;
            const int gx = x0 + lc;
            float xv = 0.0f, yv = 0.0f;
            if (gy >= 0 && gy < IMG && gx >= 0 && gx < IMG) {
                const long g = pbase + (long)gy * IMG + gx;
                xv = xin[g];
                yv = yin[g];
            }
            sx[i] = xv;
            sy[i] = yv;
        }
    }
    __syncthreads();

    // 1/|d-12| weights, 0 at center (d=12)
    constexpr float invw[25] = {
        1.f/12, 1.f/11, 1.f/10, 1.f/9,  1.f/8,
        1.f/7,  1.f/6,  1.f/5,  1.f/4,  1.f/3,
        1.f/2,  1.f/1,  0.f,    1.f/1,  1.f/2,
        1.f/3,  1.f/4,  1.f/5,  1.f/6,  1.f/7,
        1.f/8,  1.f/9,  1.f/10, 1.f/11, 1.f/12 };

    const int lx = threadIdx.x;            // 0..31 : output column within tile
    const int r0 = threadIdx.y * 4;        // base output row (4 consecutive rows/thread)

    // Stage the 8-row x 5-col window (both arrays) in registers: loaded once,
    // reused by all 4 output rows -> ~2x fewer DS instructions per pixel.
    float xv[8][5], yv[8][5];
    #pragma unroll
    for (int j = 0; j < 8; ++j) {
        #pragma unroll
        for (int c = 0; c < 5; ++c) {
            const int idx = (r0 + j) * LDIM + lx + c;
            xv[j][c] = sx[idx];
            yv[j][c] = sy[idx];
        }
    }

    // Per-row 5-wide sums of y, shared across the 4 outputs (separable box sum).
    float rs[8];
    #pragma unroll
    for (int j = 0; j < 8; ++j)
        rs[j] = ((yv[j][0] + yv[j][1]) + (yv[j][2] + yv[j][3])) + yv[j][4];

    #pragma unroll
    for (int k = 0; k < 4; ++k) {
        const float xc = xv[k + 2][2];

        float wx = 0.0f;
        #pragma unroll
        for (int di = 0; di < 5; ++di) {
            #pragma unroll
            for (int dj = 0; dj < 5; ++dj) {
                const int d = di * 5 + dj;
                if (d == 12) continue;             // skip center
                const float dx = xv[k + di][dj] - xc;
                wx = fmaf(dx * dx, invw[d], wx);
            }
        }

        const float ysum = (((rs[k] + rs[k + 1]) + (rs[k + 2] + rs[k + 3])) + rs[k + 4])
                           - yv[k + 2][2];        // exclude center tap
        const float wy = -ysum * logf(xc);

        const long o = pbase + (long)(ty * TILE + r0 + k) * IMG + (tx * TILE + lx);
        // Streaming, write-once outputs: non-temporal so they don't evict the
        // reusable x/y working set from WGP$/L2.
        __builtin_nontemporal_store(wx, &out[o]);
        __builtin_nontemporal_store(wy, &out[nelem + o]);
    }
}

extern "C" void kernel_launch(void* const* d_in, const int* in_sizes, int n_in,
                              void* d_out, int out_size, void* d_ws, size_t ws_size,
                              hipStream_t stream) {
    const float* x = (const float*)d_in[0];
    const float* y = (const float*)d_in[1];
    // d_in[2] = mask, unused by the reference computation.
    float* out = (float*)d_out;

    const unsigned long long nelem = (unsigned long long)in_sizes[0];  // B*C*W*H
    const int planes = (int)(nelem / (unsigned long long)(IMG * IMG)); // B*C = 48

    dim3 grid(IMG / TILE, IMG / TILE, planes);   // 16 x 16 x 48
    dim3 block(32, 8, 1);                        // 256 threads = 8 wave32
    hipLaunchKernelGGL(stencil_wx_wy_kernel, grid, block, 0, stream,
                       x, y, out, nelem);
}